// EdgeEncoder_overlap_10411000725575
// MI455X (gfx1250) — compile-verified
//
#include <hip/hip_runtime.h>

// EdgeEncoder overlap embedding-bag mean, factored as:
//   out[e,:] = (hist[e,0:4] @ embedding[4,128]) / max(count[e],1)
//
// Pass 1: atomic-free histogram (segment_ids sorted -> binary-search bounds)
// Pass 2: hist @ embedding via V_WMMA_F32_16X16X4_F32 (exact K=4 shape match)

typedef __attribute__((ext_vector_type(2))) float v2f;
typedef __attribute__((ext_vector_type(8))) float v8f;

#define N_EDGES 8192
#define OUT_CH  128

// ---------------------------------------------------------------------------
// Pass 1: one wave32 per edge. Uniform binary search over sorted segment_ids
// gives [start,end); lanes count token types locally; wave-reduce via shfl_xor
// (ds lane permutes); lane 0 stores the 16B hist row.
// ---------------------------------------------------------------------------
__global__ void edge_hist_kernel(const int* __restrict__ tokens,
                                 const int* __restrict__ seg,
                                 float* __restrict__ hist, // [N_EDGES][4]
                                 int T) {
  const int wave = threadIdx.x >> 5;
  const int lane = threadIdx.x & 31;
  const int e = blockIdx.x * (blockDim.x >> 5) + wave;
  if (e >= N_EDGES) return;

  // lower_bound(e) -> start; lower_bound(e+1) -> end  (uniform across wave,
  // upper tree levels hit L2/WGP$)
  int lo = 0, hi = T;
  while (lo < hi) { int mid = (lo + hi) >> 1; if (seg[mid] < e) lo = mid + 1; else hi = mid; }
  const int start = lo;
  hi = T;
  while (lo < hi) { int mid = (lo + hi) >> 1; if (seg[mid] < e + 1) lo = mid + 1; else hi = mid; }
  const int end = lo;

  float c0 = 0.f, c1 = 0.f, c2 = 0.f, c3 = 0.f;
  for (int i = start + lane; i < end; i += 32) {   // coalesced b32 loads
    const int t = tokens[i];
    c0 += (t == 0) ? 1.0f : 0.0f;
    c1 += (t == 1) ? 1.0f : 0.0f;
    c2 += (t == 2) ? 1.0f : 0.0f;
    c3 += (t == 3) ? 1.0f : 0.0f;
  }
  #pragma unroll
  for (int m = 16; m >= 1; m >>= 1) {
    c0 += __shfl_xor(c0, m, 32);
    c1 += __shfl_xor(c1, m, 32);
    c2 += __shfl_xor(c2, m, 32);
    c3 += __shfl_xor(c3, m, 32);
  }
  if (lane == 0) {
    float4 h = make_float4(c0, c1, c2, c3);
    *(float4*)(hist + (size_t)e * 4) = h;          // 16B-aligned b128 store
  }
}

// ---------------------------------------------------------------------------
// Pass 2: one wave32 per 16-edge tile. A = 16x4 hist tile, B = 4x16 chunk of
// embedding, D = 16x16 outputs; 8 chunks cover OUT_CH=128.
//
// VGPR layouts per CDNA5 ISA (32-bit, 16x16x4):
//   A (16x4):  lanes 0-15 hold M=lane, {v0,v1} = K{0,1}; lanes 16-31 = K{2,3}
//   B (4x16):  lanes 0-15 hold N=lane, {v0,v1} = K{0,1}; lanes 16-31 = K{2,3}
//   C/D:       VGPR r: lanes 0-15 -> (M=r, N=lane); lanes 16-31 -> (M=8+r)
// EXEC is all-1s throughout (no divergence; grid divides exactly).
// ---------------------------------------------------------------------------
__global__ void edge_wmma_kernel(const float* __restrict__ hist, // [N_EDGES][4]
                                 const float* __restrict__ emb,  // [4][OUT_CH]
                                 float* __restrict__ out) {      // [N_EDGES][OUT_CH]
  const int wave = threadIdx.x >> 5;
  const int lane = threadIdx.x & 31;
  const int tile = blockIdx.x * (blockDim.x >> 5) + wave; // 16 edges per tile
  const int half = lane >> 4;      // 0: K={0,1}, 1: K={2,3}
  const int l    = lane & 15;

  // A-matrix fragment
  const int row_a = tile * 16 + l;
  v2f A;
  A.x = hist[(size_t)row_a * 4 + half * 2 + 0];
  A.y = hist[(size_t)row_a * 4 + half * 2 + 1];

  // Per-row 1/max(count,1) for the D rows this lane produces (M = half*8 + r)
  float inv[8];
  #pragma unroll
  for (int r = 0; r < 8; ++r) {
    const float* h = hist + (size_t)(tile * 16 + half * 8 + r) * 4;
    const float cnt = h[0] + h[1] + h[2] + h[3];
    inv[r] = 1.0f / fmaxf(cnt, 1.0f);
  }

  #pragma unroll
  for (int chunk = 0; chunk < 8; ++chunk) {
    const int col = chunk * 16 + l;
    v2f B;
    B.x = emb[(half * 2 + 0) * OUT_CH + col];
    B.y = emb[(half * 2 + 1) * OUT_CH + col];

    v8f C = {};
    // (neg_a, A, neg_b, B, c_mod, C, reuse_a, reuse_b)
    C = __builtin_amdgcn_wmma_f32_16x16x4_f32(false, A, false, B,
                                              (short)0, C, false, false);
    #pragma unroll
    for (int r = 0; r < 8; ++r) {
      const int m = tile * 16 + half * 8 + r;
      out[(size_t)m * OUT_CH + col] = C[r] * inv[r];
    }
  }
}

// ---------------------------------------------------------------------------
// Inputs: [0] overlap_similarity f32[8192] (unused)
//         [1] overlap_length     f32[8192] (unused)
//         [2] tokens             i32[2097152]
//         [3] segment_ids        i32[2097152] (sorted)
//         [4] embedding          f32[4*128]
//         [5] n_edges            i32 scalar (== 8192, compile-time constant)
// Output: f32[8192*128]
// Workspace: hist f32[8192*4] = 128 KB
// ---------------------------------------------------------------------------
extern "C" void kernel_launch(void* const* d_in, const int* in_sizes, int n_in,
                              void* d_out, int out_size, void* d_ws, size_t ws_size,
                              hipStream_t stream) {
  const int*   tokens = (const int*)d_in[2];
  const int*   seg    = (const int*)d_in[3];
  const float* emb    = (const float*)d_in[4];
  float*       out    = (float*)d_out;
  float*       hist   = (float*)d_ws;
  const int T = in_sizes[2];

  // Pass 1: 8 waves per 256-thread block, one edge per wave -> 1024 blocks
  edge_hist_kernel<<<N_EDGES / 8, 256, 0, stream>>>(tokens, seg, hist, T);

  // Pass 2: 8 waves per block, one 16-edge tile per wave -> 64 blocks
  edge_wmma_kernel<<<(N_EDGES / 16) / 8, 256, 0, stream>>>(hist, emb, out);
}